// ConvolutionalOverlap_45372034515106
// MI455X (gfx1250) — compile-verified
//
#include <hip/hip_runtime.h>
#include <stdint.h>

// out[b,i,h,w] = w1*x[b,0,h,w-s1]*(w>=s1) + w2*x[b,0,h,w+s2]*(w+s2<W)
//   s1=(i+1)>>1, s2=(i+2)>>1  (both in [0,128]);  out (B,I,H,W)=(2,256,256,256)
//
// Store-BW bound (128 MiB out). CDNA5 plan:
//   - stage x row in LDS via GLOBAL_LOAD_ASYNC_TO_LDS (ASYNCcnt path)
//   - 128-word zero pads on both sides of the row -> completely branchless,
//     maskless inner loop; sliding ds_load windows with immediate offsets
//   - B128 coalesced stores (default RT temporal hint: output fits in 192MB
//     L2, so stores can retire at L2 rate)

#define B_ 2
#define H_ 256
#define W_ 256
#define I_ 256
#define ICHUNK 64          // i-values per block -> grid.y = 4
#define PAD 128            // max shift; zero pad each side
#define SW (PAD + W_ + PAD) // 512 words = 2 KiB LDS

__global__ __launch_bounds__(256)
void conv_overlap_shift_kernel(const float* __restrict__ x,
                               const float* __restrict__ pw1,
                               const float* __restrict__ pw2,
                               float* __restrict__ out)
{
    __shared__ float spad[SW];

    const int tid = threadIdx.x;          // 0..255 (8 wave32s)
    const int bx  = blockIdx.x;           // b*H + h
    const int b   = bx >> 8;
    const int h   = bx & (H_ - 1);
    const int i0  = blockIdx.y * ICHUNK;

    // Zero the two 128-word pads (disjoint from the async-load destination,
    // so no DS-vs-ASYNC LDS write ordering hazard).
    if (tid < PAD) spad[tid] = 0.0f;              // left pad  [0,128)
    else           spad[tid + W_] = 0.0f;         // right pad [384,512)

    // Async-copy x[b,0,h,:] (256 f32 = 1 KiB) into spad[PAD..PAD+W):
    // each of the 256 lanes moves one dword, ASYNCcnt-tracked.
    const float* row = x + ((size_t)b * H_ + h) * W_;   // block-uniform -> SGPRs
    uint32_t lds_off = (uint32_t)(uintptr_t)(&spad[0]) + (uint32_t)(PAD + tid) * 4u;
    uint32_t g_off   = (uint32_t)tid * 4u;
    asm volatile("global_load_async_to_lds_b32 %0, %1, %2"
                 :: "v"(lds_off), "v"(g_off), "s"(row)
                 : "memory");
    asm volatile("s_wait_asynccnt 0" ::: "memory");
    __syncthreads();   // also fences the pad ds_stores across waves

    const float w1 = pw1[0];
    const float w2 = pw2[0];

    // 4 concurrent i-rows; 64 lanes x float4 cover one 256-wide row.
    const int iq = tid >> 6;        // 0..3
    const int wb = (tid & 63) * 4;  // base w of this lane's float4

    const int i  = i0 + iq;         // first i handled by this thread
    const int s1 = (i + 1) >> 1;
    const int s2 = (i + 2) >> 1;

    // Sliding LDS windows: i advances by 4 per iteration => s1,s2 advance by 2.
    const float* p1 = spad + (PAD + wb) - s1;   // ds_load offset:0/4/8/12
    const float* p2 = spad + (PAD + wb) + s2;
    float* op = out + (size_t)b * (I_ * H_ * W_) + (size_t)i * (H_ * W_)
                    + (size_t)h * W_ + wb;

    #pragma unroll 4
    for (int it = 0; it < ICHUNK; it += 4) {
        const float a0 = p1[0], a1 = p1[1], a2 = p1[2], a3 = p1[3];
        const float c0 = p2[0], c1 = p2[1], c2 = p2[2], c3 = p2[3];
        float4 v;
        v.x = w1 * a0 + w2 * c0;
        v.y = w1 * a1 + w2 * c1;
        v.z = w1 * a2 + w2 * c2;
        v.w = w1 * a3 + w2 * c3;
        *reinterpret_cast<float4*>(op) = v;     // global_store_b128, coalesced
        p1 -= 2;                                 // s1 += 2
        p2 += 2;                                 // s2 += 2
        op += (size_t)4 * (H_ * W_);             // i += 4
    }
}

extern "C" void kernel_launch(void* const* d_in, const int* in_sizes, int n_in,
                              void* d_out, int out_size, void* d_ws, size_t ws_size,
                              hipStream_t stream) {
    const float* x   = (const float*)d_in[0];
    const float* pw1 = (const float*)d_in[1];
    const float* pw2 = (const float*)d_in[2];
    float* out = (float*)d_out;

    dim3 grid(B_ * H_, I_ / ICHUNK);   // (512, 4) blocks x 256 threads
    conv_overlap_shift_kernel<<<grid, 256, 0, stream>>>(x, pw1, pw2, out);
}